// SSIM_77867757077170
// MI455X (gfx1250) — compile-verified
//
#include <hip/hip_runtime.h>

typedef __attribute__((ext_vector_type(2))) float v2f;
typedef __attribute__((ext_vector_type(8))) float v8f;

#define AS1 __attribute__((address_space(1)))
#define AS3 __attribute__((address_space(3)))

#define IMG     512
#define OUTD    502                 // 512 - 11 + 1
#define TILE    16
#define TPD     32                  // ceil(502/16)
#define BATCH   128
#define NTILES  (BATCH * TPD * TPD) // 131072 workgroups
#define SSIM_C1 (0.01f * 0.01f)
#define SSIM_C2 (0.03f * 0.03f)

// Recover the 1-D Gaussian factor from the 2-D window input:
// w2d[i][j] = g[i]*g[j]  =>  g[d] = w2d[d][5] / sqrt(w2d[5][5])
__device__ __forceinline__ float gwin(const float* __restrict__ win, int d) {
    return win[d * 11 + 5] * rsqrtf(win[60]);
}

__global__ __launch_bounds__(160)
void ssim_tile_kernel(const float* __restrict__ img1,
                      const float* __restrict__ img2,
                      const float* __restrict__ win,
                      float* __restrict__ part) {
    // Five input planes: x, y, x*x, y*y, x*y (zero-padded 26x26 -> 32x32).
    // One plane per wave => A-fragment source is pure address math, no branches.
    __shared__ float planes[5][32][33];
    // Banded weight matrices: horizontal B (K x N) and vertical A (M x K)
    __shared__ float Bh[32][17];
    __shared__ float Wv[16][33];
    // Per-channel horizontal-pass result T (rows 0..31 x cols 0..15)
    __shared__ float tch[5][32][17];
    // Per-channel 16x16 conv outputs
    __shared__ float outb[5][16][17];
    __shared__ float wsum[5];

    const int tid  = threadIdx.x;
    const int bid  = blockIdx.x;
    const int imgi = bid / (TPD * TPD);
    const int tt   = bid - imgi * (TPD * TPD);
    const int oy   = (tt / TPD) * TILE;
    const int ox   = (tt - (tt / TPD) * TPD) * TILE;

    const size_t base = (size_t)imgi * (size_t)(IMG * IMG);

#if __has_builtin(__builtin_amdgcn_global_load_async_to_lds_b32)
    // ---- CDNA5 async path: global -> LDS without a VGPR round trip ----
    // Valid lanes issue async loads (ASYNCcnt); invalid region zero-filled via
    // DS stores to *disjoint* addresses, so no cross-counter ordering hazard.
    for (int idx = tid; idx < 1024; idx += 160) {
        int r = idx >> 5, c = idx & 31;
        int gy = oy + r, gx = ox + c;
        bool ok = (r < 26) && (c < 26) && (gy < IMG) && (gx < IMG);
        size_t a = base + (size_t)gy * IMG + (size_t)gx;
        if (ok) {
            __builtin_amdgcn_global_load_async_to_lds_b32(
                (AS1 int*)(img1 + a), (AS3 int*)&planes[0][r][c], 0, 0);
            __builtin_amdgcn_global_load_async_to_lds_b32(
                (AS1 int*)(img2 + a), (AS3 int*)&planes[1][r][c], 0, 0);
        } else {
            planes[0][r][c] = 0.0f;
            planes[1][r][c] = 0.0f;
        }
    }
#else
    // ---- fallback: plain guarded loads; fold products into the same pass ----
    for (int idx = tid; idx < 1024; idx += 160) {
        int r = idx >> 5, c = idx & 31;
        int gy = oy + r, gx = ox + c;
        bool ok = (r < 26) && (c < 26) && (gy < IMG) && (gx < IMG);
        size_t a = base + (size_t)gy * IMG + (size_t)gx;
        float x = ok ? img1[a] : 0.0f;
        float y = ok ? img2[a] : 0.0f;
        planes[0][r][c] = x;
        planes[1][r][c] = y;
        planes[2][r][c] = x * x;
        planes[3][r][c] = y * y;
        planes[4][r][c] = x * y;
    }
#endif

    // ---- build banded weight matrices (overlaps the in-flight async loads) ----
    for (int idx = tid; idx < 512; idx += 160) {
        {   // Bh[j][c] = g[j-c], horizontal conv as P(32x32) x Bh(32x16)
            int j = idx >> 4, c = idx & 15, d = j - c;
            Bh[j][c] = (d >= 0 && d < 11) ? gwin(win, d) : 0.0f;
        }
        {   // Wv[y][j] = g[j-y], vertical conv as Wv(16x32) x T(32x16)
            int y = idx >> 5, j = idx & 31, d = j - y;
            Wv[y][j] = (d >= 0 && d < 11) ? gwin(win, d) : 0.0f;
        }
    }

#if __has_builtin(__builtin_amdgcn_global_load_async_to_lds_b32)
#if __has_builtin(__builtin_amdgcn_s_wait_asynccnt)
    __builtin_amdgcn_s_wait_asynccnt(0);
#else
    asm volatile("s_wait_asynccnt 0x0" ::: "memory");
#endif
    __syncthreads();
    // Derive the product planes (zeros propagate into the padded region).
    for (int idx = tid; idx < 1024; idx += 160) {
        int r = idx >> 5, c = idx & 31;
        float x = planes[0][r][c];
        float y = planes[1][r][c];
        planes[2][r][c] = x * x;
        planes[3][r][c] = y * y;
        planes[4][r][c] = x * y;
    }
#endif
    __syncthreads();

    const int ch   = tid >> 5;     // wave id == channel plane
    const int lane = tid & 31;
    const int hi   = lane >> 4;    // lane half selects K sub-pair in A/B frags
    const int ml   = lane & 15;    // M (A, D rows) / N (B, D cols)
    const float* __restrict__ pl = &planes[ch][0][0];   // this wave's plane

    // ---- horizontal pass: T = P * Bh  (two 16-row M-tiles, K in 4-chunks) ----
    for (int rt = 0; rt < 2; ++rt) {
        v8f acc = {0.f, 0.f, 0.f, 0.f, 0.f, 0.f, 0.f, 0.f};
        const int Mr = rt * 16 + ml;
#pragma unroll
        for (int q = 0; q < 7; ++q) {            // K = 0..27 covers band j<=25
            const int k0 = 4 * q + 2 * hi;
            v2f a;
            a.x = pl[Mr * 33 + k0];
            a.y = pl[Mr * 33 + k0 + 1];
            v2f b;
            b.x = Bh[k0][ml];
            b.y = Bh[k0 + 1][ml];
            acc = __builtin_amdgcn_wmma_f32_16x16x4_f32(
                false, a, false, b, (short)0, acc, false, false);
        }
#pragma unroll
        for (int v = 0; v < 8; ++v)              // D: row = v + 8*hi, col = ml
            tch[ch][rt * 16 + v + 8 * hi][ml] = acc[v];
    }

    // ---- vertical pass: Out = Wv * T (wave-private LDS; DS ops are in-order) ----
    {
        v8f acc = {0.f, 0.f, 0.f, 0.f, 0.f, 0.f, 0.f, 0.f};
#pragma unroll
        for (int q = 0; q < 7; ++q) {
            const int k0 = 4 * q + 2 * hi;
            v2f a;
            a.x = Wv[ml][k0];
            a.y = Wv[ml][k0 + 1];
            v2f b;
            b.x = tch[ch][k0][ml];
            b.y = tch[ch][k0 + 1][ml];
            acc = __builtin_amdgcn_wmma_f32_16x16x4_f32(
                false, a, false, b, (short)0, acc, false, false);
        }
#pragma unroll
        for (int v = 0; v < 8; ++v)
            outb[ch][v + 8 * hi][ml] = acc[v];
    }
    __syncthreads();

    // ---- SSIM map over the valid part of this tile + reduction ----
    float lsum = 0.0f;
    for (int idx = tid; idx < 256; idx += 160) {
        int y = idx >> 4, x = idx & 15;
        if ((oy + y) < OUTD && (ox + x) < OUTD) {
            float mu1 = outb[0][y][x], mu2 = outb[1][y][x];
            float ex2 = outb[2][y][x], ey2 = outb[3][y][x], exy = outb[4][y][x];
            float mu1s = mu1 * mu1, mu2s = mu2 * mu2, m12 = mu1 * mu2;
            float sg1  = ex2 - mu1s, sg2 = ey2 - mu2s, s12 = exy - m12;
            float num = (2.0f * m12 + SSIM_C1) * (2.0f * s12 + SSIM_C2);
            float den = (mu1s + mu2s + SSIM_C1) * (sg1 + sg2 + SSIM_C2);
            lsum += num / den;
        }
    }
#pragma unroll
    for (int off = 16; off > 0; off >>= 1)
        lsum += __shfl_down(lsum, off, 32);
    if (lane == 0) wsum[ch] = lsum;
    __syncthreads();
    if (tid == 0)
        part[bid] = wsum[0] + wsum[1] + wsum[2] + wsum[3] + wsum[4];
}

__global__ __launch_bounds__(256)
void ssim_reduce_kernel(const float* __restrict__ part,
                        float* __restrict__ out, int n, float scale) {
    __shared__ float buf[256];
    float s = 0.0f;
    for (int i = threadIdx.x; i < n; i += 256) s += part[i];
    buf[threadIdx.x] = s;
    __syncthreads();
    for (int st = 128; st > 0; st >>= 1) {
        if ((int)threadIdx.x < st) buf[threadIdx.x] += buf[threadIdx.x + st];
        __syncthreads();
    }
    if (threadIdx.x == 0) out[0] = buf[0] * scale;
}

extern "C" void kernel_launch(void* const* d_in, const int* in_sizes, int n_in,
                              void* d_out, int out_size, void* d_ws, size_t ws_size,
                              hipStream_t stream) {
    (void)in_sizes; (void)n_in; (void)out_size; (void)ws_size;
    const float* img1 = (const float*)d_in[0];
    const float* img2 = (const float*)d_in[1];
    const float* win  = (const float*)d_in[2];
    float* out  = (float*)d_out;
    float* part = (float*)d_ws;   // NTILES * 4B = 512 KB of scratch

    ssim_tile_kernel<<<NTILES, 160, 0, stream>>>(img1, img2, win, part);
    const float scale = 1.0f / ((float)BATCH * (float)OUTD * (float)OUTD);
    ssim_reduce_kernel<<<1, 256, 0, stream>>>(part, out, NTILES, scale);
}